// IFNO_85117661872628
// MI455X (gfx1250) — compile-verified
//
#include <hip/hip_runtime.h>
#include <math.h>

#define B_   8
#define S_   256
#define TIN  10
#define W_   32
#define C0   33      // width + 1 input channels for block 0
#define NM   24      // kept modes per axis (12 low + 12 high)

typedef float v2f __attribute__((ext_vector_type(2)));
typedef float v8f __attribute__((ext_vector_type(8)));

__device__ __forceinline__ int kx_of(int j){ return j < 12 ? j : 232 + j; }  // 0..11, 244..255
__device__ __forceinline__ int ky_of(int j){ return j < 12 ? j : 105 + j; }  // 0..11, 117..128

__device__ __forceinline__ float gelu_exact(float v){
    return 0.5f * v * (1.0f + erff(v * 0.70710678118654752440f));
}

// f32 WMMA: D(16x16) = A(16x4) * B(4x16) + C
__device__ __forceinline__ v8f wmma4(v2f a, v2f b, v8f c){
    return __builtin_amdgcn_wmma_f32_16x16x4_f32(false, a, false, b, (short)0, c, false, false);
}

// ---------------------------------------------------------------------------
// Trig tables.
//  tabX [p][j][{cos,sin}]          : +2*pi*kx_j*p/256   (fwd2 / inv1)
//  tabYF[p][n] n=2j+ri             : ri==0 ? cos : -sin (forward DFT along y, sign folded)
//  tabI [y][n] n=2j+ri             : ri==0 ? a*cos/N2 : -a*sin/N2 (irfft weights)
// ---------------------------------------------------------------------------
__global__ void k_tables(float* __restrict__ tabX, float* __restrict__ tabYF,
                         float* __restrict__ tabI){
    int idx = blockIdx.x * blockDim.x + threadIdx.x;
    if (idx >= 256 * NM) return;
    int p = idx / NM, j = idx - p * NM;
    const float w = 6.28318530717958647692f / 256.0f;
    int kx = kx_of(j), ky = ky_of(j);
    float ax = w * (float)((kx * p) & 255);
    float ay = w * (float)((ky * p) & 255);
    tabX[idx*2 + 0] = cosf(ax);
    tabX[idx*2 + 1] = sinf(ax);
    tabYF[idx*2 + 0] =  cosf(ay);
    tabYF[idx*2 + 1] = -sinf(ay);
    float alpha = (ky == 0 || ky == 128) ? 1.0f : 2.0f;
    float s = alpha * (1.0f / 65536.0f);
    tabI[idx*2 + 0] =  s * cosf(ay);
    tabI[idx*2 + 1] = -s * sinf(ay);
}

// ---------------------------------------------------------------------------
// Lift: h[b,c,x,y] = (fc0(concat(x,gx,gy)) , a) * a  -> NCHW, 33 channels
// ---------------------------------------------------------------------------
__global__ void k_lift(const float* __restrict__ x, const float* __restrict__ ap,
                       const float* __restrict__ w, const float* __restrict__ bias,
                       float* __restrict__ h){
    int p = blockIdx.x * blockDim.x + threadIdx.x;
    if (p >= B_ * S_ * S_) return;
    int yi = p & 255, xi = (p >> 8) & 255, b = p >> 16;
    float in[12];
    #pragma unroll
    for (int t = 0; t < TIN; ++t) in[t] = x[(long)p * TIN + t];
    in[10] = (float)xi * (1.0f / 255.0f);
    in[11] = (float)yi * (1.0f / 255.0f);
    float av = ap[p];
    long baseo = (long)b * C0 * (S_*S_) + (long)xi * S_ + yi;
    for (int c = 0; c < W_; ++c){
        float s = bias[c];
        #pragma unroll
        for (int t = 0; t < 12; ++t) s = fmaf(in[t], w[t * W_ + c], s);
        h[baseo + (long)c * (S_*S_)] = s * av;
    }
    h[baseo + (long)W_ * (S_*S_)] = av * av;   // channel 32 = a_para, then *a
}

// ---------------------------------------------------------------------------
// Forward DFT along y: t1[row][n] = sum_y h[row][y]*tabYF[y][n]
// rows=(b,c,x); GEMM M x 256 x 48. One wave: 32 rows (2 M-tiles) x 48 cols,
// B-fragments shared across both M-tiles.
// ---------------------------------------------------------------------------
__global__ void k_fwd1(const float* __restrict__ h, const float* __restrict__ tabYF,
                       float* __restrict__ t1){
    int lane = threadIdx.x & 31, lo = lane & 15, hi = lane >> 4;
    long row0 = (long)blockIdx.x * 32;
    const float* hr0 = h + (row0 + lo) * S_;
    const float* hr1 = hr0 + 16 * S_;
    v8f acc[6] = {};
    for (int k0 = 0; k0 < S_; k0 += 4){
        int kk = k0 + 2*hi;
        v2f a0, a1;
        a0.x = hr0[kk]; a0.y = hr0[kk + 1];
        a1.x = hr1[kk]; a1.y = hr1[kk + 1];
        __builtin_prefetch(hr0 + kk + 64, 0, 3);   // global_prefetch_b8, stream-ahead
        #pragma unroll
        for (int nt = 0; nt < 3; ++nt){
            int n = nt*16 + lo;
            v2f bb;
            bb.x = tabYF[ kk      * 48 + n];
            bb.y = tabYF[(kk + 1) * 48 + n];
            acc[nt]     = wmma4(a0, bb, acc[nt]);
            acc[nt + 3] = wmma4(a1, bb, acc[nt + 3]);
        }
    }
    #pragma unroll
    for (int mt = 0; mt < 2; ++mt)
        #pragma unroll
        for (int nt = 0; nt < 3; ++nt)
            #pragma unroll
            for (int v = 0; v < 8; ++v)
                t1[(row0 + mt*16 + v + 8*hi) * 48 + nt*16 + lo] = acc[mt*3 + nt][v];
}

// ---------------------------------------------------------------------------
// Forward DFT along x (e^{-i}): F[b,c,kx,j] = sum_x (cx - i sx)(tr + i ti)
// ---------------------------------------------------------------------------
__global__ void k_fwd2(const float* __restrict__ t1, const float* __restrict__ tabX,
                       float* __restrict__ fm, int C){
    int idx = blockIdx.x * blockDim.x + threadIdx.x;
    if (idx >= B_ * C * NM * NM) return;
    int j = idx % NM; int r = idx / NM;
    int kxi = r % NM; int bc = r / NM;
    const float* tr = t1 + (long)bc * S_ * 48;
    float fr = 0.f, fi = 0.f;
    for (int xx = 0; xx < S_; ++xx){
        float t_r = tr[xx*48 + 2*j], t_i = tr[xx*48 + 2*j + 1];
        float cx = tabX[(xx*NM + kxi)*2], sx = tabX[(xx*NM + kxi)*2 + 1];
        fr = fmaf(cx, t_r, fr); fr = fmaf( sx, t_i, fr);
        fi = fmaf(cx, t_i, fi); fi = fmaf(-sx, t_r, fi);
    }
    fm[(long)idx*2]     = fr;
    fm[(long)idx*2 + 1] = fi;
}

// ---------------------------------------------------------------------------
// Per-mode complex channel mixing with corner-selected weights.
// rb layout: [4][C][32][12][12][2]
// ---------------------------------------------------------------------------
__global__ void k_mix(const float* __restrict__ fm, const float* __restrict__ rb,
                      float* __restrict__ gm, int C){
    int idx = blockIdx.x * blockDim.x + threadIdx.x;
    if (idx >= B_ * W_ * NM * NM) return;
    int j = idx % NM; int r = idx / NM;
    int kxi = r % NM; r /= NM;
    int o = r % W_; int b = r / W_;
    int q = (kxi >= 12 ? 1 : 0) + (j >= 12 ? 2 : 0);
    int mx = kxi % 12, my = j % 12;
    const float* wp = rb + ((((long)q * C) * W_ + o) * 12 + mx) * 24 + my * 2;
    const float* fp = fm + (((long)b * C) * NM + kxi) * (NM * 2) + j * 2;
    float gr = 0.f, gi = 0.f;
    for (int c = 0; c < C; ++c){
        float wr = wp[0], wi = wp[1];
        float fr = fp[0], fi = fp[1];
        gr += fr*wr - fi*wi;
        gi += fr*wi + fi*wr;
        wp += (long)W_ * 144 * 2;
        fp += (long)NM * NM * 2;
    }
    gm[(long)idx*2] = gr; gm[(long)idx*2 + 1] = gi;
}

// ---------------------------------------------------------------------------
// Inverse along x (e^{+i}): U[b,o,x,j] = sum_kx (cx + i sx)(gr + i gi)
// ---------------------------------------------------------------------------
__global__ void k_inv1(const float* __restrict__ gm, const float* __restrict__ tabX,
                       float* __restrict__ u1){
    int idx = blockIdx.x * blockDim.x + threadIdx.x;
    if (idx >= B_ * W_ * S_ * NM) return;
    int j = idx % NM; int r = idx / NM;
    int xx = r % S_; int bo = r / S_;
    const float* gp = gm + (long)bo * NM * NM * 2;
    float ur = 0.f, ui = 0.f;
    #pragma unroll 4
    for (int kxi = 0; kxi < NM; ++kxi){
        float gr = gp[(kxi*NM + j)*2], gi = gp[(kxi*NM + j)*2 + 1];
        float cx = tabX[(xx*NM + kxi)*2], sx = tabX[(xx*NM + kxi)*2 + 1];
        ur += cx*gr - sx*gi;
        ui += cx*gi + sx*gr;
    }
    long o = ((long)bo * S_ + xx) * 48 + 2*j;
    u1[o] = ur; u1[o + 1] = ui;
}

// ---------------------------------------------------------------------------
// Inverse along y fused with irfft weights: hout[row][y] = sum_k u1[row][k]*tabI[y][k]
// rows=(b,o,x); GEMM 65536 x 48 x 256. One wave: 32 rows x 64 cols,
// B-fragments shared across both M-tiles.
// ---------------------------------------------------------------------------
__global__ void k_inv2(const float* __restrict__ u1, const float* __restrict__ tabI,
                       float* __restrict__ hout){
    int lane = threadIdx.x & 31, lo = lane & 15, hi = lane >> 4;
    long row0 = (long)blockIdx.x * 32;
    int y0 = blockIdx.y * 64;
    const float* ur0 = u1 + (row0 + lo) * 48;
    const float* ur1 = ur0 + 16 * 48;
    v8f acc[8] = {};
    #pragma unroll
    for (int k0 = 0; k0 < 48; k0 += 4){
        int kk = k0 + 2*hi;
        v2f a0, a1;
        a0.x = ur0[kk]; a0.y = ur0[kk + 1];
        a1.x = ur1[kk]; a1.y = ur1[kk + 1];
        #pragma unroll
        for (int nt = 0; nt < 4; ++nt){
            int y = y0 + nt*16 + lo;
            v2f bb;
            bb.x = tabI[y*48 + kk];
            bb.y = tabI[y*48 + kk + 1];
            acc[nt]     = wmma4(a0, bb, acc[nt]);
            acc[nt + 4] = wmma4(a1, bb, acc[nt + 4]);
        }
    }
    #pragma unroll
    for (int mt = 0; mt < 2; ++mt)
        #pragma unroll
        for (int nt = 0; nt < 4; ++nt)
            #pragma unroll
            for (int v = 0; v < 8; ++v)
                hout[(row0 + mt*16 + v + 8*hi) * S_ + y0 + nt*16 + lo] = acc[mt*4 + nt][v];
}

// ---------------------------------------------------------------------------
// 1x1 conv skip + bias + optional GELU, accumulated onto spectral output.
// One wave per (b,x): 32 channels x 256 y. K over input channels (constexpr).
// ---------------------------------------------------------------------------
template<int C, bool GELU>
__global__ void k_conv(const float* __restrict__ hin, const float* __restrict__ w,
                       const float* __restrict__ bias, float* __restrict__ hout){
    constexpr int KS = (C + 3) / 4;
    int lane = threadIdx.x & 31, lo = lane & 15, hi = lane >> 4;
    int bx = blockIdx.x;
    int b = bx >> 8, xx = bx & 255;
    long ibase = (long)b * C  * (S_*S_) + (long)xx * S_;
    long obase = (long)b * W_ * (S_*S_) + (long)xx * S_;
    // preload weight A-fragments for both M-tiles and all K-steps
    v2f afrag[2][KS];
    #pragma unroll
    for (int mt = 0; mt < 2; ++mt)
        #pragma unroll
        for (int ks = 0; ks < KS; ++ks){
            int c0 = ks*4 + 2*hi;
            afrag[mt][ks].x = (c0     < C) ? w[(mt*16 + lo)*C + c0]     : 0.0f;
            afrag[mt][ks].y = (c0 + 1 < C) ? w[(mt*16 + lo)*C + c0 + 1] : 0.0f;
        }
    for (int nt = 0; nt < 16; ++nt){
        int y = nt*16 + lo;
        v8f acc0, acc1;
        #pragma unroll
        for (int v = 0; v < 8; ++v){
            int o = v + 8*hi;
            acc0[v] = hout[obase + (long)o        * (S_*S_) + y] + bias[o];
            acc1[v] = hout[obase + (long)(o + 16) * (S_*S_) + y] + bias[o + 16];
        }
        #pragma unroll
        for (int ks = 0; ks < KS; ++ks){
            int c0 = ks*4 + 2*hi;
            v2f bb;
            bb.x = (c0     < C) ? hin[ibase + (long)c0     * (S_*S_) + y] : 0.0f;
            bb.y = (c0 + 1 < C) ? hin[ibase + (long)(c0+1) * (S_*S_) + y] : 0.0f;
            acc0 = wmma4(afrag[0][ks], bb, acc0);
            acc1 = wmma4(afrag[1][ks], bb, acc1);
        }
        #pragma unroll
        for (int v = 0; v < 8; ++v){
            int o = v + 8*hi;
            float v0 = acc0[v], v1 = acc1[v];
            if (GELU){ v0 = gelu_exact(v0); v1 = gelu_exact(v1); }
            hout[obase + (long)o        * (S_*S_) + y] = v0;
            hout[obase + (long)(o + 16) * (S_*S_) + y] = v1;
        }
    }
}

// ---------------------------------------------------------------------------
// Head: out = gelu(h @ fc1 + b1) @ fc2 + b2. One wave per (b,x).
// fc1 via WMMA (128 x 256 x K=32 per wave), fc2 dot + cross-half shuffle.
// ---------------------------------------------------------------------------
__global__ void k_fc1(const float* __restrict__ h, const float* __restrict__ w1,
                      const float* __restrict__ b1, const float* __restrict__ w2,
                      const float* __restrict__ b2, float* __restrict__ out){
    int lane = threadIdx.x & 31, lo = lane & 15, hi = lane >> 4;
    int bx = blockIdx.x; int b = bx >> 8, xx = bx & 255;
    long ibase = (long)b * W_ * (S_*S_) + (long)xx * S_;
    float b2v = b2[0];
    for (int yt = 0; yt < 16; ++yt){
        int y = yt*16 + lo;
        v2f bf[8];
        #pragma unroll
        for (int ks = 0; ks < 8; ++ks){
            int c0 = ks*4 + 2*hi;
            bf[ks].x = h[ibase + (long)c0     * (S_*S_) + y];
            bf[ks].y = h[ibase + (long)(c0+1) * (S_*S_) + y];
        }
        float part = 0.0f;
        #pragma unroll
        for (int mt = 0; mt < 8; ++mt){
            int t0 = mt * 16;
            v8f acc;
            #pragma unroll
            for (int v = 0; v < 8; ++v) acc[v] = b1[t0 + v + 8*hi];
            #pragma unroll
            for (int ks = 0; ks < 8; ++ks){
                int c0 = ks*4 + 2*hi;
                v2f a;
                a.x = w1[ c0      * 128 + t0 + lo];
                a.y = w1[(c0 + 1) * 128 + t0 + lo];
                acc = wmma4(a, bf[ks], acc);
            }
            #pragma unroll
            for (int v = 0; v < 8; ++v)
                part = fmaf(gelu_exact(acc[v]), w2[t0 + v + 8*hi], part);
        }
        float tot = part + __shfl_xor(part, 16, 32);
        if (hi == 0)
            out[(long)bx * S_ + yt*16 + lo] = tot + b2v;
    }
}

// ---------------------------------------------------------------------------
extern "C" void kernel_launch(void* const* d_in, const int* in_sizes, int n_in,
                              void* d_out, int out_size, void* d_ws, size_t ws_size,
                              hipStream_t stream){
    (void)in_sizes; (void)n_in; (void)out_size; (void)ws_size;
    const float* x    = (const float*)d_in[0];
    const float* ap   = (const float*)d_in[1];
    const float* fc0w = (const float*)d_in[2];
    const float* fc0b = (const float*)d_in[3];
    const float* rb[4] = {(const float*)d_in[4], (const float*)d_in[5],
                          (const float*)d_in[6], (const float*)d_in[7]};
    const float* ww[4] = {(const float*)d_in[8],  (const float*)d_in[10],
                          (const float*)d_in[12], (const float*)d_in[14]};
    const float* wb[4] = {(const float*)d_in[9],  (const float*)d_in[11],
                          (const float*)d_in[13], (const float*)d_in[15]};
    const float* fc1w = (const float*)d_in[16];
    const float* fc1b = (const float*)d_in[17];
    const float* fc2w = (const float*)d_in[18];
    const float* fc2b = (const float*)d_in[19];
    float* out = (float*)d_out;

    float* ws = (float*)d_ws;
    size_t off = 0;
    float* tabX  = ws + off; off += 256*NM*2;
    float* tabYF = ws + off; off += 256*NM*2;
    float* tabI  = ws + off; off += 256*NM*2;
    float* hA    = ws + off; off += (size_t)B_*C0*S_*S_;
    float* hB    = ws + off; off += (size_t)B_*W_*S_*S_;
    float* t1    = ws + off; off += (size_t)B_*C0*S_*48;
    float* fm    = ws + off; off += (size_t)B_*C0*NM*NM*2;
    float* gm    = ws + off; off += (size_t)B_*W_*NM*NM*2;
    float* u1    = ws + off; off += (size_t)B_*W_*S_*48;

    k_tables<<<(256*NM + 63)/64, 64, 0, stream>>>(tabX, tabYF, tabI);
    k_lift<<<(B_*S_*S_)/256, 256, 0, stream>>>(x, ap, fc0w, fc0b, hA);

    for (int blk = 0; blk < 4; ++blk){
        int C = (blk == 0) ? C0 : W_;
        float* hin   = (blk & 1) ? hB : hA;
        float* houtp = (blk & 1) ? hA : hB;
        int rows = B_ * C * S_;
        k_fwd1<<<rows/32, 32, 0, stream>>>(hin, tabYF, t1);
        int t2 = B_ * C * NM * NM;
        k_fwd2<<<(t2 + 255)/256, 256, 0, stream>>>(t1, tabX, fm, C);
        int t3 = B_ * W_ * NM * NM;
        k_mix<<<(t3 + 255)/256, 256, 0, stream>>>(fm, rb[blk], gm, C);
        int t4 = B_ * W_ * S_ * NM;
        k_inv1<<<(t4 + 255)/256, 256, 0, stream>>>(gm, tabX, u1);
        k_inv2<<<dim3((B_*W_*S_)/32, 4), 32, 0, stream>>>(u1, tabI, houtp);
        if (blk == 0)      k_conv<33, true ><<<B_*S_, 32, 0, stream>>>(hin, ww[blk], wb[blk], houtp);
        else if (blk < 3)  k_conv<32, true ><<<B_*S_, 32, 0, stream>>>(hin, ww[blk], wb[blk], houtp);
        else               k_conv<32, false><<<B_*S_, 32, 0, stream>>>(hin, ww[blk], wb[blk], houtp);
    }
    // final block output landed in hA
    k_fc1<<<B_*S_, 32, 0, stream>>>(hA, fc1w, fc1b, fc2w, fc2b, out);
}